// MaskedConv2d_69569880261013
// MI455X (gfx1250) — compile-verified
//
#include <hip/hip_runtime.h>

// Reference: masked conv where the sparse path recomputes exactly the values the
// dense path produced -> output == relu(conv3x3(x, weight[perm])) with
// perm = [sorted dense channels, sorted sparse channels] from ch_mask.
// spa_mask (d_in[1]) is numerically irrelevant.

#define Hc 384
#define Wc 384
#define CIN 64
#define COUT 64
#define NPIX (Hc * Wc)

#define SW_FLOATS (9 * 64 * 64)        // [tap][m][k] permuted weights: 36864 floats
#define SX_STRIDE 72                   // row stride (floats): 16B-aligned interior, 2addr-mergeable
#define SX_FLOATS (3 * 64 * SX_STRIDE) // 13824 floats
#define SX_BYTE_BASE (SW_FLOATS * 4)   // LDS byte offset of sX (dynamic LDS starts at 0)

typedef __attribute__((ext_vector_type(2))) float v2f;
typedef __attribute__((ext_vector_type(4))) float v4f;
typedef __attribute__((ext_vector_type(8))) float v8f;
typedef __attribute__((ext_vector_type(4))) unsigned v4u;
typedef __attribute__((ext_vector_type(8))) unsigned v8u;

// CDNA5 async copy global->LDS, 16 bytes per lane, tracked by ASYNCcnt.
__device__ __forceinline__ void async_load_b128(unsigned lds_byte_off,
                                                const void* gaddr) {
  asm volatile("global_load_async_to_lds_b128 %0, %1, off"
               :
               : "v"(lds_byte_off), "v"((unsigned long long)gaddr));
}
__device__ __forceinline__ void wait_asynccnt0() {
  asm volatile("s_wait_asynccnt 0x0" ::: "memory");
}

// --- kernel 0: derive channel permutation from ch_mask, copy ch_mask to tail ---
__global__ void setup_perm_kernel(const float* __restrict__ ch_mask,
                                  int* __restrict__ perm,
                                  float* __restrict__ out_tail) {
  int t = threadIdx.x;
  if (t < 2 * COUT) out_tail[t] = ch_mask[t];  // second tuple output: ch_mask
  if (t == 0) {
    int nd = 0;
    for (int c = 0; c < COUT; ++c)             // dense channels first (ascending)
      if (ch_mask[2 * c + 1] > ch_mask[2 * c + 0]) perm[nd++] = c;
    for (int c = 0; c < COUT; ++c)             // then sparse channels (ascending)
      if (!(ch_mask[2 * c + 1] > ch_mask[2 * c + 0])) perm[nd++] = c;
  }
}

// --- kernel 0b: one-time weight permute+transpose into d_ws ---
// wout[t*4096 + m*64 + c] = weight[perm[m]][c][kh][kw], t = kh*3+kw
__global__ void weight_prep_kernel(const float* __restrict__ weight,
                                   const int* __restrict__ perm,
                                   float* __restrict__ wout) {
  int idx = blockIdx.x * 256 + threadIdx.x;
  if (idx >= SW_FLOATS) return;
  int t = idx >> 12;
  int m = (idx >> 6) & 63;
  int c = idx & 63;
  wout[idx] = weight[perm[m] * 576 + c * 9 + t];
}

// --- kernel 1: 3x3 conv as 9 shifted GEMMs with V_WMMA_F32_16X16X4_F32 ---
// Block: 256 threads (8 waves). Output tile: 64 channels x 64 pixels (one row
// segment). Wave w: channel tile (w&3)*16, pixel half (w>>2)*32.
__global__ __launch_bounds__(256, 1)
void masked_conv_wmma(const float* __restrict__ x,
                      const float* __restrict__ wsW,   // prepped weights in d_ws
                      float* __restrict__ out) {
  extern __shared__ float smem[];
  float* sW = smem;                 // [9][64][64] at LDS offset 0
  float* sX = smem + SW_FLOATS;     // [3][64][72]; cols: 3=left halo, 4..67=interior, 68=right halo

  const int tid = threadIdx.x;
  const int w0 = blockIdx.x * 64;
  const int h = blockIdx.y;

  // ---- Stage weights: one TDM bulk copy (144 KB), issued by wave 0 only. ----
  // D# (2 SGPR groups): 2D tensor of 18432 x 8-byte elements, single row.
  if (tid < 32) {
    unsigned long long ga = (unsigned long long)wsW;
    v4u g0;
    g0.x = 1u;                                                // count=1, user mode
    g0.y = 0u;                                                // lds_addr = 0 (sW base)
    g0.z = (unsigned)ga;                                      // global_addr[31:0]
    g0.w = (unsigned)((ga >> 32) & 0x01FFFFFFu) | (2u << 30); // addr[56:32], type=2
    v8u g1;
    g1.s0 = 0x00030000u;  // workgroup_mask=0, data_size=3 (8B), no flags
    g1.s1 = 0x48000000u;  // tensor_dim0 = 18432 (low 16 bits in [31:16])
    g1.s2 = 0x00010000u;  // tensor_dim0 hi = 0, tensor_dim1 = 1
    g1.s3 = 0x48000000u;  // tile_dim0 = 18432
    g1.s4 = 0x00000001u;  // tile_dim1 = 1, tile_dim2 unused
    g1.s5 = 0x00004800u;  // tensor_dim0_stride = 18432
    g1.s6 = 0u;
    g1.s7 = 0u;
    asm volatile("tensor_load_to_lds %0, %1" : : "s"(g0), "s"(g1) : "memory");
  }

  // ---- Stage x interior: per-lane async b128; zero-fill OOB rows. ----
  for (int idx = tid; idx < 3 * 64 * 16; idx += 256) {
    int g = idx & 15;
    int c = (idx >> 4) & 63;
    int kh = idx >> 10;
    int hh = h + kh - 1;
    unsigned lds_off =
        SX_BYTE_BASE + (unsigned)(((kh * 64 + c) * SX_STRIDE + 4 + g * 4) * 4);
    if ((unsigned)hh < (unsigned)Hc) {
      async_load_b128(lds_off, x + ((size_t)c * Hc + hh) * Wc + w0 + g * 4);
    } else {
      v4f z = {};
      *(v4f*)(sX + (kh * 64 + c) * SX_STRIDE + 4 + g * 4) = z;
    }
  }

  // Halo columns (scalar): j=3 <-> ww=w0-1, j=68 <-> ww=w0+64.
  for (int idx = tid; idx < 3 * 64 * 2; idx += 256) {
    int side = idx & 1;
    int c = (idx >> 1) & 63;
    int kh = idx >> 7;
    int hh = h + kh - 1;
    int ww = side ? (w0 + 64) : (w0 - 1);
    float v = 0.0f;
    if ((unsigned)hh < (unsigned)Hc && (unsigned)ww < (unsigned)Wc)
      v = x[((size_t)c * Hc + hh) * Wc + ww];
    sX[(kh * 64 + c) * SX_STRIDE + (side ? 68 : 3)] = v;
  }

  wait_asynccnt0();                          // drain this wave's async-to-LDS ops
  if (tid < 32) __builtin_amdgcn_s_wait_tensorcnt(0);  // wave 0 drains the TDM copy
  __syncthreads();                           // + DScnt wait: tile visible to all waves

  const int wave = tid >> 5;
  const int lane = tid & 31;
  const int mt = wave & 3;   // channel tile 0..3
  const int ph = wave >> 2;  // pixel half 0..1
  const int ln = lane & 15;
  const int hi = lane >> 4;

  v8f acc0 = {};
  v8f acc1 = {};

  // A 16x4 f32 layout: lanes0-15 hold M=ln, K=k0+{0,1}; lanes16-31 K=k0+{2,3}
  // -> one ds_load_b64 per lane from the k-contiguous sW row.
  const float* wrow = sW + (mt * 16 + ln) * 64 + 2 * hi;

  for (int t = 0; t < 9; ++t) {
    const int kh = t / 3;
    const int kw = t - 3 * kh;
    const float* wt = wrow + (t << 12);
    // B 4x16 f32 (mirror of C/D): VGPR0 = B[k0+2*hi][n], VGPR1 = B[k0+2*hi+1][n]
    // pixel n reads x at ww = w0+n+kw-1 -> LDS col = n+kw+3.
    const float* xb = sX + (kh * 64 + 2 * hi) * SX_STRIDE + (ph * 32 + ln + kw + 3);
#pragma unroll
    for (int kc = 0; kc < 16; ++kc) {
      v2f a = *(const v2f*)(wt + kc * 4);
      const float* xp = xb + (kc * 4) * SX_STRIDE;
      v2f b0;
      b0.x = xp[0];
      b0.y = xp[SX_STRIDE];
      v2f b1;
      b1.x = xp[16];
      b1.y = xp[SX_STRIDE + 16];
      acc0 = __builtin_amdgcn_wmma_f32_16x16x4_f32(
          false, a, false, b0, (short)0, acc0, false, false);
      acc1 = __builtin_amdgcn_wmma_f32_16x16x4_f32(
          false, a, false, b1, (short)0, acc1, false, false);
    }
  }

  // D layout: VGPR r -> M = r + 8*hi, N = ln. Fused ReLU, coalesced stores.
  const int px = w0 + ph * 32 + ln;
#pragma unroll
  for (int r = 0; r < 8; ++r) {
    int m = mt * 16 + r + 8 * hi;
    float v0 = acc0[r] > 0.0f ? acc0[r] : 0.0f;
    float v1 = acc1[r] > 0.0f ? acc1[r] : 0.0f;
    out[(size_t)m * NPIX + h * Wc + px] = v0;
    out[(size_t)m * NPIX + h * Wc + px + 16] = v1;
  }
}

extern "C" void kernel_launch(void* const* d_in, const int* in_sizes, int n_in,
                              void* d_out, int out_size, void* d_ws, size_t ws_size,
                              hipStream_t stream) {
  const float* x = (const float*)d_in[0];
  // d_in[1] = spa_mask : numerically irrelevant (sparse path reproduces dense values)
  const float* weight = (const float*)d_in[2];
  const float* ch_mask = (const float*)d_in[3];
  float* out = (float*)d_out;

  // d_ws layout: [0,256) perm int[64]; [256, 256+147456) prepped weights (16B aligned)
  int* perm = (int*)d_ws;
  float* wsW = (float*)((char*)d_ws + 256);

  setup_perm_kernel<<<1, 128, 0, stream>>>(ch_mask, perm, out + (size_t)COUT * NPIX);
  weight_prep_kernel<<<(SW_FLOATS + 255) / 256, 256, 0, stream>>>(weight, perm, wsW);

  dim3 grid(Wc / 64, Hc);
  size_t shmem = (size_t)(SW_FLOATS + SX_FLOATS) * sizeof(float);  // ~198 KB of 320 KB/WGP
  masked_conv_wmma<<<grid, 256, shmem, stream>>>(x, wsW, out);
}